// PerformerAttention_32521492365389
// MI455X (gfx1250) — compile-verified
//
#include <hip/hip_runtime.h>
#include <hip/hip_bf16.h>

#define BH 32          // B*H
#define SEQ 2048       // S
#define DIM 64         // D (= feature dim)

typedef __attribute__((ext_vector_type(2))) float v2f;
typedef __attribute__((ext_vector_type(8))) float v8f;

__device__ __forceinline__ v2f ld2(const float* p) { return *(const v2f*)p; }

// ---------------------------------------------------------------------------
// Kernel 1: out[bh, s, f] = relu( sum_d x[bh, s, d] * W[f, d] )
// One wave computes a 16-row x 64-feature tile with V_WMMA_F32_16X16X4_F32.
// A fragment (16x4 f32): lanes 0-15 hold M=lane, regs {K, K+1}; lanes 16-31
// hold {K+2, K+3}.  B fragment (4x16, B = W^T): lane l holds column N=f,
// regs {K, K+1} (high half {K+2, K+3}) -> identical row-major load pattern.
// ---------------------------------------------------------------------------
__global__ __launch_bounds__(256) void perf_featmap(const float* __restrict__ x,
                                                    const float* __restrict__ W,
                                                    float* __restrict__ out) {
    const int wid  = threadIdx.x >> 5;
    const int lane = threadIdx.x & 31;
    const int hf   = lane >> 4;      // half-wave select (K offset +2)
    const int l    = lane & 15;
    const int bh   = blockIdx.y;
    const int mbase = (blockIdx.x * 8 + wid) * 16;

    const size_t rowbase = ((size_t)bh * SEQ + mbase + l) * DIM;
    v2f a[16];
#pragma unroll
    for (int k = 0; k < 16; ++k)
        a[k] = ld2(x + rowbase + k * 4 + hf * 2);

#pragma unroll
    for (int fb = 0; fb < 4; ++fb) {
        const float* wrow = W + (size_t)(fb * 16 + l) * DIM;
        v8f acc = {};
#pragma unroll
        for (int k = 0; k < 16; ++k) {
            v2f b = ld2(wrow + k * 4 + hf * 2);
            acc = __builtin_amdgcn_wmma_f32_16x16x4_f32(
                false, a[k], false, b, (short)0, acc, false, false);
        }
        // D layout: lanes 0-15: N=lane, M=reg; lanes 16-31: N=lane-16, M=reg+8
        const size_t obase = ((size_t)bh * SEQ + mbase + hf * 8) * DIM + fb * 16 + l;
#pragma unroll
        for (int r = 0; r < 8; ++r)
            out[obase + (size_t)r * DIM] = fmaxf(acc[r], 0.0f);
    }
}

// ---------------------------------------------------------------------------
// Kernel 2: per workgroup: 16 query rows x full 2048 keys.
//   scores(16x2048) via WMMA f32 -> LDS (128 KB), row softmax in LDS,
//   attn_weights streamed to global, attn_output = v * rowsum(attn).
// ---------------------------------------------------------------------------
__global__ __launch_bounds__(256) void perf_attn(const float* __restrict__ qf,
                                                 const float* __restrict__ kf,
                                                 const float* __restrict__ v,
                                                 float* __restrict__ out0,   // [BH,S,D]
                                                 float* __restrict__ outw) { // [BH,S,S]
    __shared__ float sc[16 * SEQ];    // 131072 B of the 320 KB WGP LDS
    __shared__ float rowsum[16];

    const int wid  = threadIdx.x >> 5;
    const int lane = threadIdx.x & 31;
    const int hf   = lane >> 4;
    const int l    = lane & 15;
    const int bh   = blockIdx.y;
    const int mbase = blockIdx.x * 16;

    // Hoist the 16 A fragments (whole 16x64 qf tile) into 32 VGPRs.
    const size_t qbase = ((size_t)bh * SEQ + mbase + l) * DIM;
    v2f a[16];
#pragma unroll
    for (int k = 0; k < 16; ++k)
        a[k] = ld2(qf + qbase + k * 4 + hf * 2);

    // Each wave: 16 column tiles of 16 keys -> 16x16 wmma ops.
    for (int t = 0; t < 16; ++t) {
        const int cb = (wid * 16 + t) * 16;
        const float* krow = kf + ((size_t)bh * SEQ + cb + l) * DIM;
        v8f acc = {};
#pragma unroll
        for (int k = 0; k < 16; ++k) {
            v2f b = ld2(krow + k * 4 + hf * 2);
            acc = __builtin_amdgcn_wmma_f32_16x16x4_f32(
                false, a[k], false, b, (short)0, acc, false, false);
        }
#pragma unroll
        for (int r = 0; r < 8; ++r)
            sc[(hf * 8 + r) * SEQ + cb + l] = acc[r];
    }
    __syncthreads();

    // Softmax: wave wid owns rows {2*wid, 2*wid+1}; lane covers 64 cols as
    // 16 float4 chunks (contiguous 512 B per iteration across the wave).
    for (int rr = 0; rr < 2; ++rr) {
        const int r = wid * 2 + rr;
        float4 vals[16];
        float mx = -3.402823466e38f;
#pragma unroll
        for (int i = 0; i < 16; ++i) {
            vals[i] = *(const float4*)&sc[r * SEQ + i * 128 + lane * 4];
            mx = fmaxf(mx, fmaxf(fmaxf(vals[i].x, vals[i].y),
                                 fmaxf(vals[i].z, vals[i].w)));
        }
#pragma unroll
        for (int off = 16; off >= 1; off >>= 1)
            mx = fmaxf(mx, __shfl_xor(mx, off, 32));

        float sum = 0.0f;
#pragma unroll
        for (int i = 0; i < 16; ++i) {
            vals[i].x = __expf(vals[i].x - mx);
            vals[i].y = __expf(vals[i].y - mx);
            vals[i].z = __expf(vals[i].z - mx);
            vals[i].w = __expf(vals[i].w - mx);
            sum += (vals[i].x + vals[i].y) + (vals[i].z + vals[i].w);
        }
#pragma unroll
        for (int off = 16; off >= 1; off >>= 1)
            sum += __shfl_xor(sum, off, 32);
        const float inv = 1.0f / sum;

        float rs = 0.0f;
        const size_t obase = ((size_t)bh * SEQ + mbase + r) * (size_t)SEQ;
#pragma unroll
        for (int i = 0; i < 16; ++i) {
            float4 w4;
            w4.x = vals[i].x * inv; w4.y = vals[i].y * inv;
            w4.z = vals[i].z * inv; w4.w = vals[i].w * inv;
            rs += (w4.x + w4.y) + (w4.z + w4.w);
            *(float4*)&outw[obase + i * 128 + lane * 4] = w4;   // b128 stores
        }
#pragma unroll
        for (int off = 16; off >= 1; off >>= 1)
            rs += __shfl_xor(rs, off, 32);
        if (lane == 0) rowsum[r] = rs;
    }
    __syncthreads();

    // attn_output[b,h,s,d] = v[b,h,s,d] * rowsum(attn)[s]   (faithful to ref)
    for (int idx = threadIdx.x; idx < 16 * DIM; idx += 256) {
        const int r = idx >> 6;
        const int d = idx & 63;
        const size_t g = ((size_t)bh * SEQ + mbase + r) * DIM + d;
        out0[g] = v[g] * rowsum[r];
    }
}

extern "C" void kernel_launch(void* const* d_in, const int* in_sizes, int n_in,
                              void* d_out, int out_size, void* d_ws, size_t ws_size,
                              hipStream_t stream) {
    const float* q = (const float*)d_in[0];   // [B,H,S,D]
    const float* k = (const float*)d_in[1];   // [B,H,S,D]
    const float* v = (const float*)d_in[2];   // [B,H,S,D]
    const float* W = (const float*)d_in[3];   // [D,D]

    float* out0 = (float*)d_out;                        // attn_output [B,H,S,D]
    float* outw = out0 + (size_t)BH * SEQ * DIM;        // attn_weights [B,H,S,S]

    float* qf = (float*)d_ws;                           // 16 MB
    float* kf = qf + (size_t)BH * SEQ * DIM;            // 16 MB

    dim3 blk(256);
    perf_featmap<<<dim3(SEQ / 128, BH), blk, 0, stream>>>(q, W, qf);
    perf_featmap<<<dim3(SEQ / 128, BH), blk, 0, stream>>>(k, W, kf);
    perf_attn<<<dim3(SEQ / 16, BH), blk, 0, stream>>>(qf, kf, v, out0, outw);
}